// CVT_12446815223882
// MI455X (gfx1250) — compile-verified
//
#include <hip/hip_runtime.h>
#include <math.h>

typedef __attribute__((ext_vector_type(2))) float v2f;
typedef __attribute__((ext_vector_type(8))) float v8f;

#define BN_  12      // B*NC
#define HW_  1600    // H*W
#define W_   40
#define C_   256
#define NT_  100     // HW/16 k-tiles / n-tiles

// ---------------------------------------------------------------------------
// Kernel A: per-pixel cylindrical coordinates (theta, height)
// ---------------------------------------------------------------------------
__global__ void coords_kernel(const float* __restrict__ depth,
                              const float* __restrict__ K,
                              const float* __restrict__ T,
                              float* __restrict__ th_out,
                              float* __restrict__ z_out) {
  int p = blockIdx.x * blockDim.x + threadIdx.x;
  if (p >= BN_ * HW_) return;
  int bn  = p / HW_;
  int pix = p - bn * HW_;
  int h = pix / W_;
  int w = pix - h * W_;

  float d = depth[p];                       // (B,NC,1,H,W) flat == (BN,HW)
  const float* Kb = K + bn * 9;
  float fx = Kb[0], cx = Kb[2], fy = Kb[4], cy = Kb[5];
  float X = ((float)w - cx) / fx * d;
  float Y = ((float)h - cy) / fy * d;

  const float* Tb = T + bn * 16;            // pts_ego[i] = T[i][:] . [X,Y,d,1]
  float ex = Tb[0]*X + Tb[1]*Y + Tb[2]*d  + Tb[3];
  float ey = Tb[4]*X + Tb[5]*Y + Tb[6]*d  + Tb[7];
  float ez = Tb[8]*X + Tb[9]*Y + Tb[10]*d + Tb[11];
  bool valid = (d > 0.0f) && (d <= 500.0f);
  if (!valid) { ex = 0.0f; ey = 0.0f; ez = 0.0f; }

  float vx = ex, vy = ey, vz = ez - 1.5f;   // center = (0,0,1.5)
  float r  = sqrtf(vx*vx + vy*vy);
  float rs = (r == 0.0f) ? 1e-6f : r;
  float t  = 2.0f / rs;                     // R_FIXED / r_safe
  th_out[p] = atan2f(-t * vy, -t * vx);     // proj - center = -t*v
  z_out[p]  = -t * vz;
}

// ---------------------------------------------------------------------------
// Kernel B: v_vec = x @ Wv^T   (19200x256 @ 256x256), fp32 WMMA 16x16x4
// One wave per 16x16 output tile; 8 waves / 256-thread block.
// ---------------------------------------------------------------------------
__global__ __launch_bounds__(256) void vproj_kernel(const float* __restrict__ x,
                                                    const float* __restrict__ Wv,
                                                    float* __restrict__ vv) {
  int wave = threadIdx.x >> 5;
  int lane = threadIdx.x & 31;
  int tile = blockIdx.x * 8 + wave;         // 19200 tiles total
  int mt = tile >> 4;                       // 0..1199
  int ct = tile & 15;                       // 0..15
  int m0 = mt * 16;
  int c0 = ct * 16;
  int M    = lane & 15;                     // A-row / B-col / D-col index
  int half = lane >> 4;

  const float* arow = x  + (size_t)(m0 + M) * C_;   // A[m][k] = x[m][k]
  const float* brow = Wv + (size_t)(c0 + M) * C_;   // B[k][c] = Wv[c][k]

  v8f acc = {};
  #pragma unroll 8
  for (int kk = 0; kk < C_; kk += 4) {
    int k2 = kk + 2 * half;                 // even -> 8B-aligned v2f loads
    v2f a = *(const v2f*)(arow + k2);
    v2f b = *(const v2f*)(brow + k2);
    acc = __builtin_amdgcn_wmma_f32_16x16x4_f32(false, a, false, b,
                                                (short)0, acc, false, false);
  }
  #pragma unroll
  for (int v = 0; v < 8; ++v)
    vv[(size_t)(m0 + v + 8 * half) * C_ + c0 + M] = acc[v];
}

// ---------------------------------------------------------------------------
// Kernel C: out[b] = sim[b] @ v_vec[b], sim tile built on the fly in LDS.
// Block = 512 threads = 16 waves; block owns (batch, 16-row) stripe, wave w
// owns columns [16w, 16w+16). Per 16-wide k-step: 256 threads compute the
// 16x16 sim tile into LDS (one exp per entry, shared by all 16 waves), then
// each wave issues 4x v_wmma_f32_16x16x4_f32 with A from LDS (ds_load_b64).
// ---------------------------------------------------------------------------
__global__ __launch_bounds__(512) void fused_sim_gemm(const float* __restrict__ th,
                                                      const float* __restrict__ z,
                                                      const float* __restrict__ cov,
                                                      const float* __restrict__ vv,
                                                      float* __restrict__ out) {
  __shared__ __align__(16) float s[16 * 18];   // pitch 18 -> 8B-aligned pairs
  const int tid  = threadIdx.x;
  const int wave = tid >> 5;                   // c-tile id 0..15
  const int lane = tid & 31;
  const int bb = blockIdx.x / NT_;             // batch 0..11
  const int nt = blockIdx.x - bb * NT_;        // n-tile 0..99
  const int n0 = nt * 16;
  const int c0 = wave * 16;
  const int M    = lane & 15;
  const int half = lane >> 4;

  // inverse of 2x2 cov (general, matches jnp.linalg.inv)
  float c00 = cov[0], c01 = cov[1], c10 = cov[2], c11 = cov[3];
  float id  = 1.0f / (c00 * c11 - c01 * c10);
  float i00 = c11 * id, i11 = c00 * id;
  float i01s = -(c01 + c10) * id;

  const float* thb = th + bb * HW_;
  const float* zb  = z  + bb * HW_;

  // phase-1 mapping: thread t<256 computes sim row si, col sj
  const int si = tid >> 4, sj = tid & 15;
  float thi = 0.f, zi = 0.f;
  if (tid < 256) { thi = thb[n0 + si]; zi = zb[n0 + si]; }

  const float PI    = 3.14159265358979323846f;
  const float TWOPI = 6.28318530717958647692f;

  v8f acc = {};
  for (int m0 = 0; m0 < HW_; m0 += 16) {
    __syncthreads();                          // protect s[] from previous step
    if (tid < 256) {
      float dth = thi - thb[m0 + sj];
      float wr  = dth + PI;                   // (dth+pi) mod 2pi - pi
      wr -= floorf(wr * (1.0f / TWOPI)) * TWOPI;
      dth = wr - PI;
      float e0 = dth;                         // RADIUS==1
      float e1 = zi - zb[m0 + sj];
      float d2 = i00 * e0 * e0 + i01s * e0 * e1 + i11 * e1 * e1;
      d2 = fmaxf(d2, 0.0f);
      s[si * 18 + sj] = (d2 <= 1.44f) ? expf(-0.5f * d2) : 0.0f;  // TRUNC^2
    }
    __syncthreads();

    const float* vvb = vv + ((size_t)(bb * HW_ + m0)) * C_ + c0 + M;
    #pragma unroll
    for (int sub = 0; sub < 4; ++sub) {
      int kb = 4 * sub + 2 * half;            // A: K = kb, kb+1
      v2f a = *(const v2f*)(&s[M * 18 + kb]); // ds_load_b64
      v2f b;
      b.x = vvb[(kb + 0) * C_];               // B[k][c] = v_vec[m0+k][c]
      b.y = vvb[(kb + 1) * C_];
      acc = __builtin_amdgcn_wmma_f32_16x16x4_f32(false, a, false, b,
                                                  (short)0, acc, false, false);
    }
  }

  float* ob = out + ((size_t)(bb * HW_ + n0)) * C_ + c0 + M;
  #pragma unroll
  for (int v = 0; v < 8; ++v)
    ob[(v + 8 * half) * C_] = acc[v];
}

// ---------------------------------------------------------------------------
extern "C" void kernel_launch(void* const* d_in, const int* in_sizes, int n_in,
                              void* d_out, int out_size, void* d_ws, size_t ws_size,
                              hipStream_t stream) {
  const float* depth = (const float*)d_in[0];
  const float* K     = (const float*)d_in[1];
  const float* T     = (const float*)d_in[2];
  const float* cov   = (const float*)d_in[3];
  const float* x     = (const float*)d_in[4];
  const float* Wv    = (const float*)d_in[5];
  float* out = (float*)d_out;

  // workspace layout: th (19200) | z (19200) | v_vec (19200*256) ~ 19.8 MB
  float* th = (float*)d_ws;
  float* z  = th + BN_ * HW_;
  float* vv = z  + BN_ * HW_;

  coords_kernel<<<(BN_ * HW_ + 255) / 256, 256, 0, stream>>>(depth, K, T, th, z);
  vproj_kernel<<<(BN_ * HW_ / 16) * 16 / 8, 256, 0, stream>>>(x, Wv, vv);
  fused_sim_gemm<<<BN_ * NT_, 512, 0, stream>>>(th, z, cov, vv, out);
}